// TopKRouter_15796889715414
// MI455X (gfx1250) — compile-verified
//
#include <hip/hip_runtime.h>
#include <math.h>

typedef float v2f __attribute__((ext_vector_type(2)));
typedef float v8f __attribute__((ext_vector_type(8)));

#define D_MODEL      4096
#define NUM_EXPERTS  64
#define TOP_K        8
#define NUM_TOKENS   (4 * 4096)

#define WAVES_PER_BLOCK 4
#define TOK_PER_WAVE    32                                  // 2 M-tiles per wave
#define TOK_PER_BLOCK   (WAVES_PER_BLOCK * TOK_PER_WAVE)    // 128 tokens / block
#define BLOCK_THREADS   (WAVES_PER_BLOCK * 32)              // 128 threads

__global__ __launch_bounds__(BLOCK_THREADS)
void topk_router_kernel(const float* __restrict__ x,
                        const float* __restrict__ W,
                        float* __restrict__ w_out,
                        int*   __restrict__ idx_out,
                        float* __restrict__ p_out) {
  // logits staging: 128 tokens x 64 experts = 32 KB
  __shared__ float sm[TOK_PER_BLOCK * NUM_EXPERTS];

  const int tid      = threadIdx.x;
  const int lane     = tid & 31;
  const int wave     = tid >> 5;
  const int laneHalf = lane >> 4;       // 0: K={0,1}, 1: K={2,3}
  const int l16      = lane & 15;       // row (A) / column (B) within tile
  const int koff     = laneHalf << 1;   // 0 or 2

  const int block_tok = blockIdx.x * TOK_PER_BLOCK;
  const int wave_tok  = block_tok + wave * TOK_PER_WAVE;

  // A: lane covers token rows (wave_tok + l16) and (wave_tok + 16 + l16)
  const float* aptr = x + (size_t)(wave_tok + l16) * D_MODEL + koff;
  // B: lane covers expert column l16 (per 16-expert tile), same k-slots
  const float* bptr = W + (size_t)l16 * D_MODEL + koff;

  v8f c00 = {}, c01 = {}, c02 = {}, c03 = {};   // tokens [0,16)  x experts 0..63
  v8f c10 = {}, c11 = {}, c12 = {}, c13 = {};   // tokens [16,32) x experts 0..63

  for (int kk = 0; kk < D_MODEL; kk += 16) {
    // stream both token stripes of x ahead (~512B/lane ahead within each row)
    __builtin_prefetch(aptr + kk + 128, 0, 0);
    __builtin_prefetch(aptr + 16 * D_MODEL + kk + 128, 0, 0);
#pragma unroll
    for (int j = 0; j < 16; j += 4) {
      const int k = kk + j;
      v2f a0 = *(const v2f*)(aptr + k);
      v2f a1 = *(const v2f*)(aptr + 16 * D_MODEL + k);
      v2f b0 = *(const v2f*)(bptr + k);
      v2f b1 = *(const v2f*)(bptr + 16 * D_MODEL + k);
      v2f b2 = *(const v2f*)(bptr + 32 * D_MODEL + k);
      v2f b3 = *(const v2f*)(bptr + 48 * D_MODEL + k);
      c00 = __builtin_amdgcn_wmma_f32_16x16x4_f32(false, a0, false, b0, (short)0, c00, false, false);
      c01 = __builtin_amdgcn_wmma_f32_16x16x4_f32(false, a0, false, b1, (short)0, c01, false, false);
      c02 = __builtin_amdgcn_wmma_f32_16x16x4_f32(false, a0, false, b2, (short)0, c02, false, false);
      c03 = __builtin_amdgcn_wmma_f32_16x16x4_f32(false, a0, false, b3, (short)0, c03, false, false);
      c10 = __builtin_amdgcn_wmma_f32_16x16x4_f32(false, a1, false, b0, (short)0, c10, false, false);
      c11 = __builtin_amdgcn_wmma_f32_16x16x4_f32(false, a1, false, b1, (short)0, c11, false, false);
      c12 = __builtin_amdgcn_wmma_f32_16x16x4_f32(false, a1, false, b2, (short)0, c12, false, false);
      c13 = __builtin_amdgcn_wmma_f32_16x16x4_f32(false, a1, false, b3, (short)0, c13, false, false);
    }
  }

  // C/D layout: VGPR r holds row M=r (lanes 0-15) / M=r+8 (lanes 16-31), N = l16
  {
    float* my = sm + wave * TOK_PER_WAVE * NUM_EXPERTS;
#pragma unroll
    for (int r = 0; r < 8; ++r) {
      const int m = r + laneHalf * 8;
      my[m * NUM_EXPERTS +  0 + l16] = c00[r];
      my[m * NUM_EXPERTS + 16 + l16] = c01[r];
      my[m * NUM_EXPERTS + 32 + l16] = c02[r];
      my[m * NUM_EXPERTS + 48 + l16] = c03[r];
      my[(m + 16) * NUM_EXPERTS +  0 + l16] = c10[r];
      my[(m + 16) * NUM_EXPERTS + 16 + l16] = c11[r];
      my[(m + 16) * NUM_EXPERTS + 32 + l16] = c12[r];
      my[(m + 16) * NUM_EXPERTS + 48 + l16] = c13[r];
    }
  }
  __syncthreads();

  // softmax: one thread per token (all 128 threads active)
  {
    float* vals = sm + tid * NUM_EXPERTS;
    float mx = vals[0];
    for (int e = 1; e < NUM_EXPERTS; ++e) mx = fmaxf(mx, vals[e]);
    float s = 0.0f;
    for (int e = 0; e < NUM_EXPERTS; ++e) {
      float ex = __expf(vals[e] - mx);
      vals[e] = ex;
      s += ex;
    }
    const float inv = 1.0f / s;
    for (int e = 0; e < NUM_EXPERTS; ++e) vals[e] *= inv;
  }
  __syncthreads();

  // coalesced router_probs writeback (sm is token-major, matches global layout)
  {
    const float4* src = (const float4*)sm;
    float4* dst = (float4*)(p_out + (size_t)block_tok * NUM_EXPERTS);
    for (int i = tid; i < TOK_PER_BLOCK * NUM_EXPERTS / 4; i += BLOCK_THREADS)
      dst[i] = src[i];
  }
  __syncthreads();  // probs read out before top-k mutates sm

  // top-8 + renormalize: one thread per token
  {
    float* vals = sm + tid * NUM_EXPERTS;
    const int token = block_tok + tid;
    float wsel[TOP_K];
    int   isel[TOP_K];
    float wsum = 0.0f;
#pragma unroll
    for (int j = 0; j < TOP_K; ++j) {
      float best = -1.0f;
      int   bi = 0;
      for (int e = 0; e < NUM_EXPERTS; ++e) {
        const float v = vals[e];
        if (v > best) { best = v; bi = e; }   // strict '>' => lowest index on ties
      }
      vals[bi] = -1.0f;                        // probs >= 0, safe sentinel
      wsel[j] = best;
      isel[j] = bi;
      wsum += best;
    }
    const float winv = 1.0f / (wsum + 1e-9f);
#pragma unroll
    for (int j = 0; j < TOP_K; ++j) {
      w_out[(size_t)token * TOP_K + j]   = wsel[j] * winv;
      idx_out[(size_t)token * TOP_K + j] = isel[j];
    }
  }
}

extern "C" void kernel_launch(void* const* d_in, const int* in_sizes, int n_in,
                              void* d_out, int out_size, void* d_ws, size_t ws_size,
                              hipStream_t stream) {
  (void)in_sizes; (void)n_in; (void)out_size; (void)d_ws; (void)ws_size;
  const float* x = (const float*)d_in[0];
  const float* W = (const float*)d_in[1];

  float* out     = (float*)d_out;
  float* w_out   = out;                                        // [16384, 8] f32
  int*   idx_out = (int*)(out + (size_t)NUM_TOKENS * TOP_K);   // [16384, 8] i32
  float* p_out   = out + (size_t)NUM_TOKENS * TOP_K * 2;       // [16384, 64] f32

  dim3 grid(NUM_TOKENS / TOK_PER_BLOCK);   // 128 blocks
  dim3 block(BLOCK_THREADS);               // 128 threads = 4 waves
  hipLaunchKernelGGL(topk_router_kernel, grid, block, 0, stream,
                     x, W, w_out, idx_out, p_out);
}